// NoveltyGNN_43233140801786
// MI455X (gfx1250) — compile-verified
//
#include <hip/hip_runtime.h>
#include <math.h>

typedef __attribute__((ext_vector_type(2))) float v2f;
typedef __attribute__((ext_vector_type(8))) float v8f;

#define N_NODES_C 20000
#define NUM_GRAPHS_C 64

// ---------------- elementwise helpers ----------------

__global__ void fill_kernel(float* __restrict__ p, float v, int n) {
  int i = blockIdx.x * blockDim.x + threadIdx.x;
  if (i < n) p[i] = v;
}

__global__ void deg_kernel(const int* __restrict__ dst, float* __restrict__ deg, int E) {
  int e = blockIdx.x * blockDim.x + threadIdx.x;
  if (e < E) atomicAdd(&deg[dst[e]], 1.0f);
}

__global__ void rsqrt_kernel(float* __restrict__ deg, int n) {
  int i = blockIdx.x * blockDim.x + threadIdx.x;
  if (i < n) deg[i] = rsqrtf(deg[i]);   // deg >= 1 always (self loop)
}

// ---------------- WMMA f32 GEMM: C[M,N] = A[M,K] * W[K,N] ----------------
// Register-blocked: one wave computes a 32x64 output block = 2 M-tiles x 4
// N-tiles of v_wmma_f32_16x16x4_f32 (8 v8f accumulators). Per k-step: 2 A
// fragment loads + 4 B fragment loads feed 8 WMMAs, giving 4x A reuse and
// 2x B reuse vs the naive wave-per-tile version (all operands L2-resident).
// Fragment layouts per CDNA5 ISA 7.12.2:
//   A 16x4 : lanes 0-15 hold (M=lane, K=0/1) in v0/v1; lanes 16-31 hold K=2/3.
//   B 4x16 : lanes 0-15 hold (K=0/1, N=lane) in v0/v1; lanes 16-31 hold K=2/3.
//   C 16x16: VGPR v, lanes 0-15 -> (M=v, N=lane); lanes 16-31 -> (M=v+8, N=lane-16).
__global__ void __launch_bounds__(128)
wmma_gemm_f32(const float* __restrict__ A, const float* __restrict__ W,
              float* __restrict__ C, int M, int K, int N) {
  const int wave = threadIdx.x >> 5;
  const int lane = threadIdx.x & 31;
  const int mgroups = M >> 5;                   // 32-row groups
  const int ngroups = N >> 6;                   // 64-col groups
  const int t = blockIdx.x * 4 + wave;
  if (t >= mgroups * ngroups) return;           // wave-uniform guard (EXEC stays all-1s)
  const int tm = t / ngroups;
  const int tn = t - tm * ngroups;

  const int row   = lane & 15;                  // A row / B col within tile
  const int khalf = lane >> 4;                  // which K-pair this half-wave holds

  const float* __restrict__ A0 = A + (size_t)(tm * 32 + row) * K;
  const float* __restrict__ A1 = A0 + (size_t)16 * K;
  const float* __restrict__ B0 = W + (size_t)(tn * 64 + row);

  v8f acc[2][4] = {};
  for (int kk = 0; kk < K; kk += 4) {
    const int kb = kk + khalf * 2;
    v2f a0, a1;
    a0.x = A0[kb];
    a0.y = A0[kb + 1];
    a1.x = A1[kb];
    a1.y = A1[kb + 1];
    const float* __restrict__ Brow0 = B0 + (size_t)kb * N;
    const float* __restrict__ Brow1 = B0 + (size_t)(kb + 1) * N;
#pragma unroll
    for (int j = 0; j < 4; ++j) {
      v2f b;
      b.x = Brow0[j * 16];
      b.y = Brow1[j * 16];
      acc[0][j] = __builtin_amdgcn_wmma_f32_16x16x4_f32(
          false, a0, false, b, (short)0, acc[0][j], false, false);
      acc[1][j] = __builtin_amdgcn_wmma_f32_16x16x4_f32(
          false, a1, false, b, (short)0, acc[1][j], false, false);
    }
  }

  const int rbase = khalf ? 8 : 0;
#pragma unroll
  for (int mi = 0; mi < 2; ++mi) {
#pragma unroll
    for (int j = 0; j < 4; ++j) {
      float* __restrict__ Cout =
          C + (size_t)(tm * 32 + mi * 16) * N + tn * 64 + j * 16 + (lane & 15);
#pragma unroll
      for (int v = 0; v < 8; ++v) {
        Cout[(size_t)(v + rbase) * N] = acc[mi][j][v];
      }
    }
  }
}

// ---------------- edge scatter: AGG[dst] += H[src] * dinv[src]*dinv[dst] ----------------
// One thread handles a (edge, 4-channel) pair; gathered float4 read + 4 f32 atomics.
// All buffers are L2-resident (192 MB global L2), so atomics resolve in L2.
__global__ void scatter_kernel(const float* __restrict__ H, const int* __restrict__ src,
                               const int* __restrict__ dst, const float* __restrict__ dinv,
                               float* __restrict__ AGG, int E, int C, int qshift) {
  int idx = blockIdx.x * blockDim.x + threadIdx.x;
  int e = idx >> qshift;
  if (e >= E) return;
  int q = idx & ((1 << qshift) - 1);
  int s = src[e];
  int d = dst[e];
  float norm = dinv[s] * dinv[d];
  float4 h = *((const float4*)(H + (size_t)s * C) + q);
  float* out = AGG + (size_t)d * C + (q << 2);
  atomicAdd(out + 0, h.x * norm);
  atomicAdd(out + 1, h.y * norm);
  atomicAdd(out + 2, h.z * norm);
  atomicAdd(out + 3, h.w * norm);
}

// ---------------- fused self-loop + bias + ReLU (in place on AGG) ----------------
__global__ void fuse_kernel(const float* __restrict__ H, const float* __restrict__ dinv,
                            const float* __restrict__ bias, float* __restrict__ AGG,
                            int total, int cshift) {
  int idx = blockIdx.x * blockDim.x + threadIdx.x;
  if (idx >= total) return;
  int n = idx >> cshift;
  int c = idx & ((1 << cshift) - 1);
  float di = dinv[n];
  float v = AGG[idx] + H[idx] * di * di + bias[c];
  AGG[idx] = fmaxf(v, 0.0f);
}

// ---------------- global mean pool (accumulate) ----------------
__global__ void pool_kernel(const float* __restrict__ H, const int* __restrict__ batch,
                            float* __restrict__ sums, float* __restrict__ cnt,
                            int total, int cshift) {
  int idx = blockIdx.x * blockDim.x + threadIdx.x;
  if (idx >= total) return;
  int n = idx >> cshift;
  int c = idx & ((1 << cshift) - 1);
  int b = batch[n];
  atomicAdd(&sums[(b << cshift) + c], H[idx]);
  if (c == 0) atomicAdd(&cnt[b], 1.0f);
}

// ---------------- MLP head: relu(g@Wl1+bl1) @ Wl2 + bl2 -> sigmoid ----------------
__global__ void head_kernel(const float* __restrict__ sums, const float* __restrict__ cnt,
                            const float* __restrict__ Wl1, const float* __restrict__ bl1,
                            const float* __restrict__ Wl2, const float* __restrict__ bl2,
                            float* __restrict__ out) {
  int g = threadIdx.x;
  if (g >= NUM_GRAPHS_C) return;
  float inv = 1.0f / fmaxf(cnt[g], 1.0f);
  float a[64];
#pragma unroll 4
  for (int j = 0; j < 64; ++j) {
    float s = bl1[j];
    for (int c = 0; c < 128; ++c) {
      s += (sums[g * 128 + c] * inv) * Wl1[c * 64 + j];
    }
    a[j] = fmaxf(s, 0.0f);
  }
  float o = bl2[0];
#pragma unroll
  for (int j = 0; j < 64; ++j) o += a[j] * Wl2[j];
  out[g] = 1.0f / (1.0f + expf(-o));
}

// ---------------- launcher ----------------

extern "C" void kernel_launch(void* const* d_in, const int* in_sizes, int n_in,
                              void* d_out, int out_size, void* d_ws, size_t ws_size,
                              hipStream_t stream) {
  const float* x   = (const float*)d_in[0];
  const int* ei    = (const int*)d_in[1];
  const int* batch = (const int*)d_in[2];
  const float* W1  = (const float*)d_in[3];
  const float* b1  = (const float*)d_in[4];
  const float* W2  = (const float*)d_in[5];
  const float* b2  = (const float*)d_in[6];
  const float* Wl1 = (const float*)d_in[7];
  const float* bl1 = (const float*)d_in[8];
  const float* Wl2 = (const float*)d_in[9];
  const float* bl2 = (const float*)d_in[10];
  float* out = (float*)d_out;

  const int N = in_sizes[2];          // 20000 nodes
  const int E = in_sizes[1] / 2;      // 320000 edges
  const int* src = ei;
  const int* dst = ei + E;

  // workspace layout (all 256B aligned):
  //   [0)           dinv        N floats
  //   [81920)       pool        64*128 sums + 64 counts
  //   [131072)      bufA        N*256 f32  (H1; later H2 in lower half, AGG2 in upper half)
  //   [+20480000)   bufB        N*256 f32  (AGG1 -> relu'd layer-1 output)
  char* w = (char*)d_ws;
  float* dinv = (float*)(w);
  float* sums = (float*)(w + 81920);
  float* cnt  = sums + NUM_GRAPHS_C * 128;
  float* bufA = (float*)(w + 131072);
  float* bufB = (float*)(w + 131072 + (size_t)N_NODES_C * 256 * 4);
  float* H1   = bufA;
  float* AGG1 = bufB;
  float* H2   = bufA;                         // reuse (H1 dead after fuse1)
  float* AGG2 = bufA + (size_t)N * 128;       // upper half of bufA

  const int TB = 256;

  // 1) degree (self loop counts as 1) -> dinv
  fill_kernel<<<(N + TB - 1) / TB, TB, 0, stream>>>(dinv, 1.0f, N);
  deg_kernel<<<(E + TB - 1) / TB, TB, 0, stream>>>(dst, dinv, E);
  rsqrt_kernel<<<(N + TB - 1) / TB, TB, 0, stream>>>(dinv, N);

  // 2) layer 1: H1 = x @ W1 ; AGG1 = scatter(norm * H1) ; relu(AGG1 + self + b1)
  fill_kernel<<<((N * 256) + TB - 1) / TB, TB, 0, stream>>>(AGG1, 0.0f, N * 256);
  {
    int waves = (N >> 5) * (256 >> 6);        // 625 * 4 = 2500 waves of 32x64 blocks
    wmma_gemm_f32<<<(waves + 3) / 4, 128, 0, stream>>>(x, W1, H1, N, 256, 256);
  }
  {
    int total = E << 6;                       // E * 64 float4 jobs
    scatter_kernel<<<(total + TB - 1) / TB, TB, 0, stream>>>(H1, src, dst, dinv, AGG1, E, 256, 6);
  }
  fuse_kernel<<<((N * 256) + TB - 1) / TB, TB, 0, stream>>>(H1, dinv, b1, AGG1, N * 256, 8);

  // 3) layer 2: H2 = AGG1 @ W2 ; AGG2 = scatter ; relu(+self +b2)
  {
    int waves = (N >> 5) * (128 >> 6);        // 625 * 2 = 1250 waves
    wmma_gemm_f32<<<(waves + 3) / 4, 128, 0, stream>>>(AGG1, W2, H2, N, 256, 128);
  }
  fill_kernel<<<((N * 128) + TB - 1) / TB, TB, 0, stream>>>(AGG2, 0.0f, N * 128);
  {
    int total = E << 5;                       // E * 32 float4 jobs
    scatter_kernel<<<(total + TB - 1) / TB, TB, 0, stream>>>(H2, src, dst, dinv, AGG2, E, 128, 5);
  }
  fuse_kernel<<<((N * 128) + TB - 1) / TB, TB, 0, stream>>>(H2, dinv, b2, AGG2, N * 128, 7);

  // 4) mean pool + head
  fill_kernel<<<((NUM_GRAPHS_C * 128 + NUM_GRAPHS_C) + TB - 1) / TB, TB, 0, stream>>>(
      sums, 0.0f, NUM_GRAPHS_C * 128 + NUM_GRAPHS_C);
  {
    int total = N << 7;
    pool_kernel<<<(total + TB - 1) / TB, TB, 0, stream>>>(AGG2, batch, sums, cnt, total, 7);
  }
  head_kernel<<<1, 64, 0, stream>>>(sums, cnt, Wl1, bl1, Wl2, bl2, out);
}